// HungarianMatcher_7559142441270
// MI455X (gfx1250) — compile-verified
//
#include <hip/hip_runtime.h>
#include <hip/hip_bf16.h>

// Problem constants (match reference)
#define B_  64
#define Q_  900
#define T_  100
#define C_  256
#define TT_ 7                                // ceil(T/16)
#define INF_ 1e9f

typedef __attribute__((ext_vector_type(2))) float v2f;
typedef __attribute__((ext_vector_type(8))) float v8f;

// ---------------------------------------------------------------------------
// Kernel 1: cost matrix via WMMA (class term) + VALU epilogue (L1 + GIoU).
// One wave32 computes a full 16(Q) x 100(T) stripe: each A fragment (logits)
// is loaded ONCE and multiplied into 7 accumulators (one per 16-wide T tile),
// cutting logits HBM traffic 7x (to the 59 MB minimum).
//   cost_class = logits @ (-onehot(labels)) -> exact in f32 WMMA.
// ---------------------------------------------------------------------------
__global__ __launch_bounds__(256) void cost_matrix_kernel(
    const float* __restrict__ pred_logits,   // [B,Q,C]
    const float* __restrict__ pred_boxes,    // [B,Q,4] cxcywh
    const int*   __restrict__ tgt_labels,    // [B,T]
    const float* __restrict__ tgt_boxes,     // [B,T,4] cxcywh
    float*       __restrict__ out_cost,      // [B,Q,T]
    float*       __restrict__ out_costT)     // [B,T,Q] in d_ws (may be null)
{
  const int TQ = (Q_ + 15) / 16;             // 57
  // grid is sized so every wave has work: B*TQ = 3648 = 456 blocks * 8 waves
  const int wave = blockIdx.x * (blockDim.x >> 5) + (threadIdx.x >> 5);
  const int b  = wave / TQ;
  const int tq = wave - b * TQ;
  const int q0 = tq * 16;

  const int lane = threadIdx.x & 31;
  const int n    = lane & 15;                // A row / B col / D col index
  const int half = lane >> 4;                // K pair base = 2*half

  // ---- B fragment sources: labels of the 7 target column groups.
  //      All guards are selects (no branches) -> EXEC stays all-ones.
  int lbl[TT_];
  #pragma unroll
  for (int tt = 0; tt < TT_; ++tt) {
    const int t   = tt * 16 + n;
    const int tcl = (t < T_) ? t : (T_ - 1);
    int l = tgt_labels[(size_t)b * T_ + tcl];
    lbl[tt] = (t < T_) ? l : -1;             // -1 never matches k in [0,256)
  }

  // ---- A fragment source: row M = lane&15, columns k + 2*half.
  const int qa  = q0 + n;
  const int qac = (qa < Q_) ? qa : (Q_ - 1);
  const float* arow =
      pred_logits + ((size_t)b * Q_ + (size_t)qac) * C_ + 2 * half;

  v8f acc[TT_];
  #pragma unroll
  for (int tt = 0; tt < TT_; ++tt)
    acc[tt] = (v8f){0.f, 0.f, 0.f, 0.f, 0.f, 0.f, 0.f, 0.f};

  for (int k = 0; k < C_; k += 4) {
    if ((k & 63) == 0)                       // uniform in k -> no EXEC change
      __builtin_prefetch(arow + k + 64, 0, 0);  // global_prefetch_b8
    const v2f a = *(const v2f*)(arow + k);   // logits[M][k+2h], [k+2h+1]
    const int kb = k + 2 * half;             // B rows held by this lane
    #pragma unroll
    for (int tt = 0; tt < TT_; ++tt) {
      v2f bf;
      bf[0] = (lbl[tt] == kb)     ? -1.0f : 0.0f;   // -onehot
      bf[1] = (lbl[tt] == kb + 1) ? -1.0f : 0.0f;
      acc[tt] = __builtin_amdgcn_wmma_f32_16x16x4_f32(
          false, a, false, bf, (short)0, acc[tt], false, false);
    }
  }

  // ---- Epilogue: L1 + (-GIoU) per element, then store both layouts.
  // C/D layout: VGPR r -> row M = r + 8*half, lane -> col N = lane&15.
  #pragma unroll
  for (int tt = 0; tt < TT_; ++tt) {
    const int t   = tt * 16 + n;
    const int tcl = (t < T_) ? t : (T_ - 1);
    const float4 tb = ((const float4*)tgt_boxes)[(size_t)b * T_ + tcl];
    const float tbx0 = tb.x - 0.5f * tb.z, tby0 = tb.y - 0.5f * tb.w;
    const float tbx1 = tb.x + 0.5f * tb.z, tby1 = tb.y + 0.5f * tb.w;
    const float area_b = (tbx1 - tbx0) * (tby1 - tby0);

    #pragma unroll
    for (int r = 0; r < 8; ++r) {
      const int q  = q0 + r + 8 * half;
      const int qc = (q < Q_) ? q : (Q_ - 1);
      const float4 pb = ((const float4*)pred_boxes)[(size_t)b * Q_ + qc];

      const float l1 = fabsf(pb.x - tb.x) + fabsf(pb.y - tb.y) +
                       fabsf(pb.z - tb.z) + fabsf(pb.w - tb.w);

      const float px0 = pb.x - 0.5f * pb.z, py0 = pb.y - 0.5f * pb.w;
      const float px1 = pb.x + 0.5f * pb.z, py1 = pb.y + 0.5f * pb.w;
      const float area_a = (px1 - px0) * (py1 - py0);

      const float iw = fmaxf(fminf(px1, tbx1) - fmaxf(px0, tbx0), 0.f);
      const float ih = fmaxf(fminf(py1, tby1) - fmaxf(py0, tby0), 0.f);
      const float inter = iw * ih;
      const float uni   = area_a + area_b - inter;
      const float iou   = inter / uni;

      const float ew = fmaxf(fmaxf(px1, tbx1) - fminf(px0, tbx0), 0.f);
      const float eh = fmaxf(fmaxf(py1, tby1) - fminf(py0, tby0), 0.f);
      const float enc = ew * eh;
      const float giou = iou - (enc - uni) / enc;

      const float c = acc[tt][r] + l1 - giou;  // all cost weights = 1
      if (q < Q_ && t < T_) {
        out_cost[(size_t)b * Q_ * T_ + (size_t)q * T_ + t] = c;
        if (out_costT)
          out_costT[(size_t)b * T_ * Q_ + (size_t)t * Q_ + q] = c;
      }
    }
  }
}

// ---------------------------------------------------------------------------
// Kernel 2: exact JV Hungarian on cost^T [T,Q] per batch.
// One block per batch; O(Q) slack scans / dual updates parallelized across
// 256 threads with shared-memory argmin (first-index tie-break = jnp.argmin).
// Prefers the transposed [B,T,Q] copy in d_ws for contiguous scan reads.
// ---------------------------------------------------------------------------
__global__ __launch_bounds__(256) void hungarian_kernel(
    const float* __restrict__ cost,          // [B,Q,T]
    const float* __restrict__ costT,         // [B,T,Q] or null
    float* __restrict__ pred_idx,            // [B,T]
    float* __restrict__ tgt_idx)             // [B,T]
{
  const int b   = blockIdx.x;
  const int tid = threadIdx.x;
  const float* cb  = cost + (size_t)b * Q_ * T_;
  const float* cbT = costT ? (costT + (size_t)b * T_ * Q_) : nullptr;

  __shared__ float u[T_ + 1];
  __shared__ float v[Q_ + 1];
  __shared__ float minv[Q_];
  __shared__ int   p[Q_ + 1];
  __shared__ int   way[Q_];
  __shared__ unsigned char used[Q_ + 1];
  __shared__ float redv[256];
  __shared__ int   redi[256];
  __shared__ int   s_j0;
  __shared__ int   s_col[T_];

  for (int j = tid; j <= Q_; j += 256) { v[j] = 0.f; p[j] = -1; }
  for (int i = tid; i <= T_; i += 256) u[i] = 0.f;
  __syncthreads();

  for (int i = 0; i < T_; ++i) {
    for (int j = tid; j < Q_; j += 256) { minv[j] = INF_; way[j] = 0; used[j] = 0; }
    if (tid == 0) { used[Q_] = 0; p[Q_] = i; s_j0 = Q_; }
    __syncthreads();

    while (p[s_j0] >= 0) {                   // uniform: shared state after barrier
      const int j0 = s_j0;
      if (tid == 0) used[j0] = 1;
      __syncthreads();

      const int   i0  = p[j0];
      const float ui0 = u[i0];

      float lbest = INF_;
      int   lidx  = Q_;
      if (cbT) {                             // coalesced, L2-resident row scan
        const float* crow = cbT + (size_t)i0 * Q_;
        for (int j = tid; j < Q_; j += 256) {
          if (!used[j]) {
            const float cur = crow[j] - ui0 - v[j];
            if (cur < minv[j]) { minv[j] = cur; way[j] = j0; }
            const float m = minv[j];
            if (m < lbest) { lbest = m; lidx = j; }
          }
        }
      } else {                               // fallback: strided column reads
        for (int j = tid; j < Q_; j += 256) {
          if (!used[j]) {
            const float cur = cb[(size_t)j * T_ + i0] - ui0 - v[j];
            if (cur < minv[j]) { minv[j] = cur; way[j] = j0; }
            const float m = minv[j];
            if (m < lbest) { lbest = m; lidx = j; }
          }
        }
      }
      redv[tid] = lbest; redi[tid] = lidx;
      __syncthreads();
      for (int s = 128; s > 0; s >>= 1) {
        if (tid < s) {
          const float ov = redv[tid + s];
          const int   oi = redi[tid + s];
          if (ov < redv[tid] || (ov == redv[tid] && oi < redi[tid])) {
            redv[tid] = ov; redi[tid] = oi;
          }
        }
        __syncthreads();
      }
      const float delta = redv[0];
      const int   j1    = redi[0];

      // Dual updates: used columns map to distinct rows -> race-free.
      for (int j = tid; j <= Q_; j += 256) {
        if (used[j]) {
          u[p[j]] += delta;
          if (j < Q_) v[j] -= delta;
        } else if (j < Q_) {
          minv[j] -= delta;
        }
      }
      if (tid == 0) s_j0 = j1;
      __syncthreads();
    }

    if (tid == 0) {                          // augment along alternating tree
      int j0 = s_j0;
      while (j0 != Q_) {
        const int j1 = way[j0];
        p[j0] = p[j1];
        j0 = j1;
      }
    }
    __syncthreads();
  }

  // col[i] = query matched to target i
  for (int j = tid; j < Q_; j += 256) {
    const int r = p[j];
    if (r >= 0) s_col[r] = j;
  }
  __syncthreads();

  // argsort(col) via rank counting (values distinct); emit scipy-style pairs.
  for (int t = tid; t < T_; t += 256) {
    const int c = s_col[t];
    int rank = 0;
    for (int t2 = 0; t2 < T_; ++t2) rank += (s_col[t2] < c) ? 1 : 0;
    pred_idx[(size_t)b * T_ + rank] = (float)c;
    tgt_idx [(size_t)b * T_ + rank] = (float)t;
  }
}

// ---------------------------------------------------------------------------
extern "C" void kernel_launch(void* const* d_in, const int* in_sizes, int n_in,
                              void* d_out, int out_size, void* d_ws, size_t ws_size,
                              hipStream_t stream) {
  const float* pred_logits = (const float*)d_in[0];  // [B,Q,C] f32
  const float* pred_boxes  = (const float*)d_in[1];  // [B,Q,4] f32
  const int*   tgt_labels  = (const int*)  d_in[2];  // [B,T]   i32
  const float* tgt_boxes   = (const float*)d_in[3];  // [B,T,4] f32

  float* out_cost = (float*)d_out;                        // [B,Q,T]
  float* out_pred = out_cost + (size_t)B_ * Q_ * T_;      // [B,T]
  float* out_tgt  = out_pred + (size_t)B_ * T_;           // [B,T]

  // Transposed cost copy in workspace (if it fits) for coalesced JV scans.
  const size_t tbytes = (size_t)B_ * Q_ * T_ * sizeof(float);
  float* costT = (ws_size >= tbytes) ? (float*)d_ws : nullptr;

  const int TQ = (Q_ + 15) / 16;
  const int total_waves = B_ * TQ;                        // 3648, divisible by 8
  const int grid1 = (total_waves + 7) / 8;                // 456 blocks x 8 waves

  cost_matrix_kernel<<<grid1, 256, 0, stream>>>(
      pred_logits, pred_boxes, tgt_labels, tgt_boxes, out_cost, costT);

  hungarian_kernel<<<B_, 256, 0, stream>>>(out_cost, costT, out_pred, out_tgt);
}